// Conv_68693706932655
// MI455X (gfx1250) — compile-verified
//
#include <hip/hip_runtime.h>
#include <hip/hip_bf16.h>
#include <stdint.h>

typedef __attribute__((ext_vector_type(2))) float v2f;
typedef __attribute__((ext_vector_type(8))) float v8f;

static __device__ __forceinline__ unsigned int ord_encode(float f) {
  unsigned int u = __float_as_uint(f);
  return u ^ ((u >> 31) ? 0xFFFFFFFFu : 0x80000000u);
}
static __device__ __forceinline__ float ord_decode(unsigned int u) {
  u ^= (u & 0x80000000u) ? 0x80000000u : 0xFFFFFFFFu;
  return __uint_as_float(u);
}

__global__ void init_kernel(unsigned int* __restrict__ m_ord,
                            float* __restrict__ ssum, int n) {
  int i = blockIdx.x * blockDim.x + threadIdx.x;
  if (i < n) { m_ord[i] = 0u; ssum[i] = 0.0f; }
}

// key_n[n,d] = feat[n,d] * sum_k W_n[n,d,k] + b_n[n,d]   (flat index i = n*16+d)
__global__ void node_key_kernel(const float* __restrict__ feat,
                                const float* __restrict__ Wn,
                                const float* __restrict__ bn,
                                float* __restrict__ key_n, int total) {
  int i = blockIdx.x * blockDim.x + threadIdx.x;
  if (i >= total) return;
  const float f = feat[i];
  const float* w = Wn + (size_t)i * 16;
  float s = 0.0f;
#pragma unroll
  for (int k = 0; k < 16; ++k) s = fmaf(f, w[k], s);
  key_n[i] = s + bn[i];
}

// One wave per edge: y = W_e @ feat[src] via 4x V_WMMA_F32_16X16X4_F32
// (x broadcast to all 16 B columns), then logit = (y + b_e + x + key_n[src]) . q[dst]
__global__ void __launch_bounds__(256)
edge_logits_kernel(const float* __restrict__ feat,
                   const float* __restrict__ key_n,
                   const float* __restrict__ query,
                   const float* __restrict__ We,
                   const float* __restrict__ be,
                   const int* __restrict__ src,
                   const int* __restrict__ dst,
                   float* __restrict__ logits,
                   unsigned int* __restrict__ m_ord,
                   int n_edges) {
  const int lane = threadIdx.x & 31;
  const int wave = threadIdx.x >> 5;
  const int e = blockIdx.x * (blockDim.x >> 5) + wave;
  if (e >= n_edges) return;           // wave-uniform: EXEC stays all-1s for WMMA
  const int hi  = lane >> 4;          // lane half selects K pair (A) / M offset (D)
  const int row = lane & 15;          // A-matrix row M

  const int sn = src[e];
  const int dn = dst[e];
  const float* W  = We    + (size_t)e  * 256;
  const float* x  = feat  + (size_t)sn * 16;
  const float* q  = query + (size_t)dn * 16;
  const float* bv = be    + (size_t)e  * 16;
  const float* kn = key_n + (size_t)sn * 16;

  // Stream-ahead: cover the full next 1KB W block (32 lanes x 32B).
  if (e + 1 < n_edges)
    __builtin_prefetch(We + (size_t)(e + 1) * 256 + (size_t)lane * 8u, 0, 0);

  v8f acc = {};
#pragma unroll
  for (int c = 0; c < 4; ++c) {
    const int k0 = 4 * c + 2 * hi;    // A VGPR0/1 hold K=k0,k0+1 per ISA layout
    // W is a 512MB single-use stream (>> 192MB L2): load non-temporal so the
    // reusable gather arrays (feat/query/key_n) stay L2-resident.
    const v2f av = __builtin_nontemporal_load((const v2f*)(W + row * 16 + k0));
    v2f bvv;
    bvv.x = x[k0];                    // broadcast x across all 16 columns
    bvv.y = x[k0 + 1];
    acc = __builtin_amdgcn_wmma_f32_16x16x4_f32(
        /*neg_a=*/false, av, /*neg_b=*/false, bvv,
        /*c_mod=*/(short)0, acc, /*reuse_a=*/false, /*reuse_b=*/false);
  }

  // D layout: VGPR r holds y[r + 8*hi] (identical across each lane half).
  float p = 0.0f;
#pragma unroll
  for (int r = 0; r < 8; ++r) {
    const int d = 8 * hi + r;
    p += (acc[r] + bv[d] + x[d] + kn[d]) * q[d];
  }
  const float logit = p + __shfl_xor(p, 16, 32);  // lo half + hi half
  if (lane == 0) {
    logits[e] = logit;
    atomicMax(&m_ord[dn], ord_encode(logit));     // segment max (ordered uint)
  }
}

__global__ void exp_sum_kernel(const float* __restrict__ logits,
                               const int* __restrict__ dst,
                               const unsigned int* __restrict__ m_ord,
                               float* __restrict__ ssum, int n_edges) {
  int e = blockIdx.x * blockDim.x + threadIdx.x;
  if (e >= n_edges) return;
  int dn = dst[e];
  float m = ord_decode(m_ord[dn]);
  atomicAdd(&ssum[dn], expf(logits[e] - m));
}

__global__ void attn_kernel(const float* __restrict__ logits,
                            const int* __restrict__ dst,
                            const unsigned int* __restrict__ m_ord,
                            const float* __restrict__ ssum,
                            float* __restrict__ attn, int n_edges) {
  int e = blockIdx.x * blockDim.x + threadIdx.x;
  if (e >= n_edges) return;
  int dn = dst[e];
  float m = ord_decode(m_ord[dn]);
  attn[e] = expf(logits[e] - m) / ssum[dn];
}

extern "C" void kernel_launch(void* const* d_in, const int* in_sizes, int n_in,
                              void* d_out, int out_size, void* d_ws, size_t ws_size,
                              hipStream_t stream) {
  const float* feat = (const float*)d_in[0];
  const float* Wn   = (const float*)d_in[1];
  const float* bn   = (const float*)d_in[2];
  const float* qry  = (const float*)d_in[3];
  const float* We   = (const float*)d_in[4];
  const float* be   = (const float*)d_in[5];
  const int*   src  = (const int*)d_in[6];
  const int*   dst  = (const int*)d_in[7];
  float* attn = (float*)d_out;

  const int D = 16;
  const int n_nodes = in_sizes[0] / D;
  const int n_edges = in_sizes[7];

  // Workspace layout (~5.6 MB): key_n | logits | m_ord | ssum
  char* ws = (char*)d_ws;
  size_t off = 0;
  float*        key_n  = (float*)(ws + off); off += (size_t)n_nodes * D * sizeof(float);
  float*        logits = (float*)(ws + off); off += (size_t)n_edges * sizeof(float);
  unsigned int* m_ord  = (unsigned int*)(ws + off); off += (size_t)n_nodes * sizeof(unsigned int);
  float*        ssum   = (float*)(ws + off); off += (size_t)n_nodes * sizeof(float);
  (void)ws_size; (void)n_in; (void)out_size;

  const int B = 256;                      // 8 wave32s per block
  init_kernel<<<(n_nodes + B - 1) / B, B, 0, stream>>>(m_ord, ssum, n_nodes);
  node_key_kernel<<<(n_nodes * D + B - 1) / B, B, 0, stream>>>(feat, Wn, bn, key_n,
                                                               n_nodes * D);
  const int wpb = B / 32;
  edge_logits_kernel<<<(n_edges + wpb - 1) / wpb, B, 0, stream>>>(
      feat, key_n, qry, We, be, src, dst, logits, m_ord, n_edges);
  exp_sum_kernel<<<(n_edges + B - 1) / B, B, 0, stream>>>(logits, dst, m_ord, ssum,
                                                          n_edges);
  attn_kernel<<<(n_edges + B - 1) / B, B, 0, stream>>>(logits, dst, m_ord, ssum, attn,
                                                       n_edges);
}